// GNNLayerPyG_12257836663487
// MI455X (gfx1250) — compile-verified
//
#include <hip/hip_runtime.h>
#include <hip/hip_bf16.h>

#define N_NODES 100000
#define N_EDGES 1000000
#define IN_F    64
#define HEADS   4
#define OUT_C   32
#define HC      128   // HEADS*OUT_C
#define EDGE_DIM 16
#define NEG_SLOPE 0.2f

typedef __attribute__((ext_vector_type(2))) float v2f;
typedef __attribute__((ext_vector_type(8))) float v8f;

// ---------- order-preserving float <-> uint map for atomicMax ----------
__device__ __forceinline__ unsigned map_ord(float f) {
  unsigned b = __float_as_uint(f);
  return (b & 0x80000000u) ? ~b : (b | 0x80000000u);
}
__device__ __forceinline__ float unmap_ord(unsigned k) {
  unsigned b = (k & 0x80000000u) ? (k & 0x7FFFFFFFu) : ~k;
  return __uint_as_float(b);
}

// ---------- 0: init out=bias, segmax=0, denom=0 ----------
__global__ void gat_init(float* __restrict__ out, const float* __restrict__ bias,
                         unsigned* __restrict__ segmax, float* __restrict__ denom) {
  int i = blockIdx.x * blockDim.x + threadIdx.x;
  if (i < N_NODES * HC) out[i] = bias[i & (HC - 1)];
  if (i < N_NODES * HEADS) { segmax[i] = 0u; denom[i] = 0.0f; }
}

// ---------- 1: x_proj = x @ W via V_WMMA_F32_16X16X4_F32 ----------
// One wave handles a 16-row tile x all 128 cols (8 N-tiles of 16).
__global__ __launch_bounds__(256) void gat_xproj_wmma(const float* __restrict__ x,
                                                      const float* __restrict__ W,
                                                      float* __restrict__ xproj) {
  const int lane = threadIdx.x & 31;
  const int wave = threadIdx.x >> 5;
  const int tile = blockIdx.x * 8 + wave;          // 16-row tile index
  if (tile >= N_NODES / 16) return;                // 100000 % 16 == 0
  const int row0 = tile * 16;
  const int m = lane & 15;                         // M within tile (A/B/C N index)
  const int khalf = (lane >> 4) << 1;              // 0 for lanes 0-15, 2 for 16-31

  v8f acc[8];
  const v8f z = {0.f,0.f,0.f,0.f,0.f,0.f,0.f,0.f};
#pragma unroll
  for (int t = 0; t < 8; ++t) acc[t] = z;

#pragma unroll
  for (int ks = 0; ks < IN_F / 4; ++ks) {
    const int kb = ks * 4 + khalf;
    // A fragment: 16x4 f32 -> lanes 0-15 K={0,1}, lanes 16-31 K={2,3}
    v2f a = *(const v2f*)(x + (row0 + m) * IN_F + kb);
#pragma unroll
    for (int t = 0; t < 8; ++t) {
      const int col = t * 16 + m;
      // B fragment: 4x16 f32 -> VGPR0 K=khalf row, VGPR1 K=khalf+1 row
      v2f b;
      b.x = W[(kb    ) * HC + col];
      b.y = W[(kb + 1) * HC + col];
      acc[t] = __builtin_amdgcn_wmma_f32_16x16x4_f32(
          false, a, false, b, (short)0, acc[t], false, false);
    }
  }
  // C/D layout: VGPR r: lanes 0-15 -> M=r, lanes 16-31 -> M=r+8; N = lane&15
  const int mofs = (lane >> 4) << 3;
#pragma unroll
  for (int t = 0; t < 8; ++t)
#pragma unroll
    for (int r = 0; r < 8; ++r)
      xproj[(row0 + r + mofs) * HC + t * 16 + m] = acc[t][r];
}

// ---------- 2: a_src/a_dst per (node, head) ----------
__global__ void gat_node_attn(const float* __restrict__ xproj,
                              const float* __restrict__ att_src,
                              const float* __restrict__ att_dst,
                              float* __restrict__ a_src, float* __restrict__ a_dst) {
  int i = blockIdx.x * blockDim.x + threadIdx.x;
  if (i >= N_NODES * HEADS) return;
  const int n = i >> 2, h = i & 3;
  const float* xp = xproj + n * HC + h * OUT_C;
  const float* as = att_src + h * OUT_C;
  const float* ad = att_dst + h * OUT_C;
  float s = 0.f, d = 0.f;
#pragma unroll
  for (int c = 0; c < OUT_C; ++c) { float v = xp[c]; s += v * as[c]; d += v * ad[c]; }
  a_src[i] = s; a_dst[i] = d;
}

// ---------- 3: fold W_edge with att_edge -> v[16][4] ----------
__global__ void gat_fold_edge(const float* __restrict__ W_edge,
                              const float* __restrict__ att_edge,
                              float* __restrict__ v) {
  int i = threadIdx.x;                  // 64 threads: i = d*4 + h
  if (i >= EDGE_DIM * HEADS) return;
  const int d = i >> 2, h = i & 3;
  float s = 0.f;
#pragma unroll
  for (int c = 0; c < OUT_C; ++c) s += W_edge[d * HC + h * OUT_C + c] * att_edge[h * OUT_C + c];
  v[i] = s;
}

// ---------- 4: edge pass 1: alpha + atomic segmax ----------
__global__ void gat_edge_alpha(const float* __restrict__ edge_attr,
                               const int* __restrict__ ei,
                               const float* __restrict__ a_src,
                               const float* __restrict__ a_dst,
                               const float* __restrict__ v,
                               float* __restrict__ alpha,
                               unsigned* __restrict__ segmax) {
  int e = blockIdx.x * blockDim.x + threadIdx.x;
  if (e >= N_EDGES) return;
  const int s = ei[e];
  const int d = ei[N_EDGES + e];
  float ae[HEADS] = {0.f, 0.f, 0.f, 0.f};
  const float4* ea4 = (const float4*)(edge_attr + e * EDGE_DIM);
#pragma unroll
  for (int q = 0; q < EDGE_DIM / 4; ++q) {
    float4 ea = ea4[q];
    const float* vq = v + q * 16;
#pragma unroll
    for (int h = 0; h < HEADS; ++h)
      ae[h] += ea.x * vq[0 * 4 + h] + ea.y * vq[1 * 4 + h] + ea.z * vq[2 * 4 + h] + ea.w * vq[3 * 4 + h];
  }
#pragma unroll
  for (int h = 0; h < HEADS; ++h) {
    float al = ae[h] + a_src[s * HEADS + h] + a_dst[d * HEADS + h];
    al = (al > 0.f) ? al : al * NEG_SLOPE;          // leaky_relu
    alpha[e * HEADS + h] = al;
    atomicMax(&segmax[d * HEADS + h], map_ord(al));
  }
}

// ---------- 5: edge pass 2: denom += exp(alpha - max) ----------
__global__ void gat_edge_denom(const float* __restrict__ alpha,
                               const int* __restrict__ ei,
                               const unsigned* __restrict__ segmax,
                               float* __restrict__ denom) {
  int e = blockIdx.x * blockDim.x + threadIdx.x;
  if (e >= N_EDGES) return;
  const int d = ei[N_EDGES + e];
  float4 al = *(const float4*)(alpha + e * HEADS);
  uint4  mk = *(const uint4*)(segmax + d * HEADS);
  atomicAdd(&denom[d * HEADS + 0], __expf(al.x - unmap_ord(mk.x)));
  atomicAdd(&denom[d * HEADS + 1], __expf(al.y - unmap_ord(mk.y)));
  atomicAdd(&denom[d * HEADS + 2], __expf(al.z - unmap_ord(mk.z)));
  atomicAdd(&denom[d * HEADS + 3], __expf(al.w - unmap_ord(mk.w)));
}

// ---------- 6: edge pass 3: normalize + gather xproj[src] + scatter-add out[dst] ----------
// 32 lanes per edge; lane l covers channels h*32 + l for h=0..3 (coalesced 128B per head).
__global__ __launch_bounds__(256) void gat_edge_scatter(const float* __restrict__ alpha,
                                                        const int* __restrict__ ei,
                                                        const unsigned* __restrict__ segmax,
                                                        const float* __restrict__ denom,
                                                        const float* __restrict__ xproj,
                                                        float* __restrict__ out) {
  const int e = blockIdx.x * 8 + (threadIdx.x >> 5);
  if (e >= N_EDGES) return;
  const int lane = threadIdx.x & 31;
  const int s = ei[e];
  const int d = ei[N_EDGES + e];
  float4 al = *(const float4*)(alpha + e * HEADS);
  uint4  mk = *(const uint4*)(segmax + d * HEADS);
  float4 dn = *(const float4*)(denom + d * HEADS);
  float w[HEADS];
  w[0] = __expf(al.x - unmap_ord(mk.x)) / (dn.x + 1e-16f);
  w[1] = __expf(al.y - unmap_ord(mk.y)) / (dn.y + 1e-16f);
  w[2] = __expf(al.z - unmap_ord(mk.z)) / (dn.z + 1e-16f);
  w[3] = __expf(al.w - unmap_ord(mk.w)) / (dn.w + 1e-16f);
  const float* xs = xproj + s * HC;
  float* od = out + d * HC;
#pragma unroll
  for (int h = 0; h < HEADS; ++h) {
    const int c = h * OUT_C + lane;
    atomicAdd(&od[c], xs[c] * w[h]);
  }
}

extern "C" void kernel_launch(void* const* d_in, const int* in_sizes, int n_in,
                              void* d_out, int out_size, void* d_ws, size_t ws_size,
                              hipStream_t stream) {
  const float* x         = (const float*)d_in[0];
  const int*   ei        = (const int*)  d_in[1];   // [2, E] flat
  const float* edge_attr = (const float*)d_in[2];
  const float* W         = (const float*)d_in[3];
  const float* W_edge    = (const float*)d_in[4];
  const float* att_src   = (const float*)d_in[5];
  const float* att_dst   = (const float*)d_in[6];
  const float* att_edge  = (const float*)d_in[7];
  const float* bias      = (const float*)d_in[8];
  float* out = (float*)d_out;

  // workspace layout (all 16B aligned)
  char* ws = (char*)d_ws;
  float*    xproj  = (float*)ws;                                    ws += (size_t)N_NODES * HC * 4;       // 51.2 MB
  float*    a_src  = (float*)ws;                                    ws += (size_t)N_NODES * HEADS * 4;    // 1.6 MB
  float*    a_dst  = (float*)ws;                                    ws += (size_t)N_NODES * HEADS * 4;    // 1.6 MB
  float*    alpha  = (float*)ws;                                    ws += (size_t)N_EDGES * HEADS * 4;    // 16 MB
  unsigned* segmax = (unsigned*)ws;                                 ws += (size_t)N_NODES * HEADS * 4;    // 1.6 MB
  float*    denom  = (float*)ws;                                    ws += (size_t)N_NODES * HEADS * 4;    // 1.6 MB
  float*    vfold  = (float*)ws;                                                                          // 256 B

  // 0: init (out = bias, segmax/denom = 0)
  {
    int total = N_NODES * HC;
    gat_init<<<(total + 255) / 256, 256, 0, stream>>>(out, bias, segmax, denom);
  }
  // 1: x_proj = x @ W  (WMMA f32)
  {
    int tiles = N_NODES / 16;               // 6250
    gat_xproj_wmma<<<(tiles + 7) / 8, 256, 0, stream>>>(x, W, xproj);
  }
  // 2: node attention coefficients
  {
    int total = N_NODES * HEADS;
    gat_node_attn<<<(total + 255) / 256, 256, 0, stream>>>(xproj, att_src, att_dst, a_src, a_dst);
  }
  // 3: fold edge attention
  gat_fold_edge<<<1, 64, 0, stream>>>(W_edge, att_edge, vfold);
  // 4: edge alpha + segment max
  gat_edge_alpha<<<(N_EDGES + 255) / 256, 256, 0, stream>>>(edge_attr, ei, a_src, a_dst, vfold, alpha, segmax);
  // 5: softmax denominators
  gat_edge_denom<<<(N_EDGES + 255) / 256, 256, 0, stream>>>(alpha, ei, segmax, denom);
  // 6: normalize + scatter messages
  gat_edge_scatter<<<(N_EDGES + 7) / 8, 256, 0, stream>>>(alpha, ei, segmax, denom, xproj, out);
}